// FKAConvNetwork_21345987461188
// MI455X (gfx1250) — compile-verified
//
#include <hip/hip_runtime.h>
#include <hip/hip_bf16.h>
#include <stdint.h>

typedef float v2f __attribute__((ext_vector_type(2)));
typedef float v4f __attribute__((ext_vector_type(4)));
typedef float v8f __attribute__((ext_vector_type(8)));

#define KNN 16
static const int NB = 2;      // batch
static const int N0 = 8192;   // base point count / pos row stride

// Async global->LDS direct copy path (CDNA5), guarded so the fallback is the
// plain VGPR-staged copy if the toolchain lacks the builtin.
#if defined(__AMDGCN__) && __has_builtin(__builtin_amdgcn_global_load_async_to_lds_b32)
#define HAVE_ASYNC_LDS 1
#else
#define HAVE_ASYNC_LDS 0
#endif
#if defined(__AMDGCN__) && __has_builtin(__builtin_amdgcn_s_wait_asynccnt)
#define WAIT_ASYNC() __builtin_amdgcn_s_wait_asynccnt(0)
#else
#define WAIT_ASYNC() ((void)0)
#endif

// ---------------------------------------------------------------------------
// f32 WMMA 16x16 tile, K accumulated 4 at a time (V_WMMA_F32_16X16X4_F32)
// ---------------------------------------------------------------------------
__device__ inline v8f wmma_f32_k4(v2f a, v2f b, v8f c) {
  return __builtin_amdgcn_wmma_f32_16x16x4_f32(false, a, false, b, (short)0, c,
                                               false, false);
}

// ---------------------------------------------------------------------------
// [B,N,3] -> [B,3,N]
// ---------------------------------------------------------------------------
__global__ void transpose_k(const float* __restrict__ in, float* __restrict__ out,
                            int N) {
  int t = blockIdx.x * blockDim.x + threadIdx.x;
  if (t >= NB * N) return;
  int b = t / N, n = t % N;
  for (int d = 0; d < 3; ++d)
    out[((size_t)b * 3 + d) * N + n] = in[((size_t)b * N + n) * 3 + d];
}

// ---------------------------------------------------------------------------
// Fused pairwise-distance + top-16 (never materializes the 537MB d2 matrix).
// pos tiles staged through LDS (async direct-to-LDS when available);
// register-resident insertion sort.
// ---------------------------------------------------------------------------
__global__ void knn_k(const float* __restrict__ posT, int n0, int N, int S,
                      float r2, int* __restrict__ idx) {
  __shared__ float sx[256], sy[256], sz[256];
  int b = blockIdx.y;
  int s = blockIdx.x * blockDim.x + threadIdx.x;
  const float* px = posT + (size_t)b * 3 * n0;
  const float* py = px + n0;
  const float* pz = py + n0;
  bool act = (s < S);
  float qx = 0.f, qy = 0.f, qz = 0.f;
  if (act) { qx = px[s]; qy = py[s]; qz = pz[s]; }
  float bd[KNN]; int bi[KNN];
#pragma unroll
  for (int q = 0; q < KNN; ++q) { bd[q] = 3.4e38f; bi[q] = 0; }
  for (int t0 = 0; t0 < N; t0 += 256) {
    int j = t0 + threadIdx.x;
    if (j < N) {
#if HAVE_ASYNC_LDS
      typedef __attribute__((address_space(1))) int gint_t;
      typedef __attribute__((address_space(3))) int lint_t;
      __builtin_amdgcn_global_load_async_to_lds_b32(
          (gint_t*)&px[j], (lint_t*)&sx[threadIdx.x], 0, 0);
      __builtin_amdgcn_global_load_async_to_lds_b32(
          (gint_t*)&py[j], (lint_t*)&sy[threadIdx.x], 0, 0);
      __builtin_amdgcn_global_load_async_to_lds_b32(
          (gint_t*)&pz[j], (lint_t*)&sz[threadIdx.x], 0, 0);
#else
      sx[threadIdx.x] = px[j]; sy[threadIdx.x] = py[j]; sz[threadIdx.x] = pz[j];
#endif
    }
    if (t0 + 256 < N) __builtin_prefetch(&px[t0 + 256 + threadIdx.x], 0, 1);
#if HAVE_ASYNC_LDS
    WAIT_ASYNC();
#endif
    __syncthreads();
    int m = N - t0; if (m > 256) m = 256;
    if (act) {
      for (int jj = 0; jj < m; ++jj) {
        float dx = sx[jj] - qx, dy = sy[jj] - qy, dz = sz[jj] - qz;
        float d2 = dx * dx + dy * dy + dz * dz;
        if (d2 < bd[KNN - 1]) {
          float d = d2; int ii = t0 + jj;
#pragma unroll
          for (int q = 0; q < KNN; ++q) {
            if (d < bd[q]) { float td = bd[q]; int ti = bi[q]; bd[q] = d; bi[q] = ii; d = td; ii = ti; }
          }
        }
      }
    }
    __syncthreads();
  }
  if (act) {
    int i0 = bi[0];
#pragma unroll
    for (int q = 0; q < KNN; ++q)
      idx[((size_t)b * S + s) * KNN + q] = (bd[q] <= r2) ? bi[q] : i0;
  }
}

// ---------------------------------------------------------------------------
// 1x1 conv as WMMA GEMM: Y[Cout,S] = W[Cout,Cin] @ X[Cin,S]  (+bias,+BN,+ReLU)
// flags: 1=bias, 2=bn, 4=relu.  One wave per 16x16 output tile.
// A (weight row) loaded as one b128 per 4-K chunk.
// ---------------------------------------------------------------------------
__global__ void conv1_wmma_k(const float* __restrict__ W, const float* __restrict__ bias,
                             const float* __restrict__ bnS, const float* __restrict__ bnB,
                             int flags, const float* __restrict__ X,
                             float* __restrict__ Y, int Cin, int Cout, int S) {
  int lane = threadIdx.x & 31;
  int half = lane >> 4, p = lane & 15;
  int sBase = blockIdx.x * 16, cBase = blockIdx.y * 16, b = blockIdx.z;
  const float* Xb = X + (size_t)b * Cin * S;
  const float* Wrow = W + (size_t)(cBase + p) * Cin;
  v8f acc = {0.f, 0.f, 0.f, 0.f, 0.f, 0.f, 0.f, 0.f};
  int off = half ? 2 : 0;
  for (int k0 = 0; k0 < Cin; k0 += 4) {
    v4f w4 = *(const v4f*)(Wrow + k0);
    int ka = k0 + off, kb = ka + 1;
    v2f A = half ? (v2f){w4.z, w4.w} : (v2f){w4.x, w4.y};
    v2f Bv = {Xb[(size_t)ka * S + sBase + p], Xb[(size_t)kb * S + sBase + p]};
    acc = wmma_f32_k4(A, Bv, acc);
  }
  float* Yb = Y + (size_t)b * Cout * S;
#pragma unroll
  for (int dv = 0; dv < 8; ++dv) {
    int c = cBase + half * 8 + dv;
    float v = acc[dv];
    if (flags & 1) v += bias[c];
    if (flags & 2) v = v * bnS[c] + bnB[c];
    if (flags & 4) v = fmaxf(v, 0.f);
    Yb[(size_t)c * S + sBase + p] = v;
  }
}

// ---------------------------------------------------------------------------
// FKAConv geometry: local coords pts/nr [B,3,S*16] and normalized dw [B,S*16]
// ---------------------------------------------------------------------------
__global__ void geom_k(const float* __restrict__ posT, int n0,
                       const int* __restrict__ idx, int S,
                       const float* aP, const float* bP, const float* nrP,
                       float* __restrict__ pts, float* __restrict__ dw) {
  int t = blockIdx.x * blockDim.x + threadIdx.x;
  if (t >= NB * S) return;
  int b = t / S, s = t % S;
  float alpha = *aP, beta = *bP, nr = *nrP;
  const float* px = posT + (size_t)b * 3 * n0;
  float qx = px[s], qy = px[(size_t)n0 + s], qz = px[2 * (size_t)n0 + s];
  int S16 = S * 16;
  float dwv[KNN]; float sum = 0.f;
#pragma unroll
  for (int k = 0; k < KNN; ++k) {
    int n = idx[((size_t)b * S + s) * KNN + k];
    float dx = px[n] - qx, dy = px[(size_t)n0 + n] - qy, dz = px[2 * (size_t)n0 + n] - qz;
    float dist = sqrtf(dx * dx + dy * dy + dz * dz);
    size_t base = (size_t)b * 3 * S16 + (size_t)s * 16 + k;
    pts[base] = dx / nr;
    pts[base + S16] = dy / nr;
    pts[base + 2 * (size_t)S16] = dz / nr;
    float w = 1.f / (1.f + expf(alpha * dist / nr - beta));
    dwv[k] = w; sum += w;
  }
  sum = sum + (sum == 0.f ? 1.f : 0.f) + 1e-6f;
  float invs = (float)KNN / sum;
#pragma unroll
  for (int k = 0; k < KNN; ++k)
    dw[(size_t)b * S16 + (size_t)s * 16 + k] = dwv[k] * invs;
}

// ---------------------------------------------------------------------------
// Kernel-MLP stage: Z[16, S*16] = Wm[16,Kdim] @ Bsrc[Kdim, S*16]
// mode 0: Bsrc = pts (Kdim=3, zero-padded to 4)
// mode 1: Bsrc = concat(mat[16,S*16], broadcast mp[16,S]) (Kdim=32)
// epi 1:  Z = relu(Z) * dw  (fc3);  epi 0: raw (for instance-norm stats)
// ---------------------------------------------------------------------------
__global__ void mlp_wmma_k(const float* __restrict__ Wm, int Kdim,
                           const float* __restrict__ Bm, const float* __restrict__ Bmp,
                           int mode, const float* __restrict__ dw, int epi,
                           float* __restrict__ Z, int S) {
  int lane = threadIdx.x & 31;
  int half = lane >> 4, p = lane & 15;
  int b = blockIdx.z;
  int S16 = S * 16;
  int col = blockIdx.x * 16 + p;
  auto loadB = [&](int r) -> float {
    if (mode == 0) return Bm[((size_t)b * 3 + r) * S16 + col];
    if (r < 16) return Bm[((size_t)b * 16 + r) * S16 + col];
    return Bmp[((size_t)b * 16 + (r - 16)) * S + (col >> 4)];
  };
  v8f acc = {0.f, 0.f, 0.f, 0.f, 0.f, 0.f, 0.f, 0.f};
  int off = half ? 2 : 0;
  int Kpad = (Kdim + 3) & ~3;
  for (int k0 = 0; k0 < Kpad; k0 += 4) {
    int ka = k0 + off, kb = ka + 1;
    float a0 = 0.f, a1 = 0.f, b0 = 0.f, b1 = 0.f;
    if (ka < Kdim) { a0 = Wm[p * Kdim + ka]; b0 = loadB(ka); }
    if (kb < Kdim) { a1 = Wm[p * Kdim + kb]; b1 = loadB(kb); }
    v2f A = {a0, a1};
    v2f Bv = {b0, b1};
    acc = wmma_f32_k4(A, Bv, acc);
  }
#pragma unroll
  for (int dv = 0; dv < 8; ++dv) {
    int c = half * 8 + dv;
    float v = acc[dv];
    if (epi == 1) v = fmaxf(v, 0.f) * dw[(size_t)b * S16 + col];
    Z[((size_t)b * 16 + c) * S16 + col] = v;
  }
}

// ---------------------------------------------------------------------------
// InstanceNorm stats: per (b,c) sum & sumsq over S*16 (LDS tree + atomicAdd)
// ---------------------------------------------------------------------------
__global__ void zero_k(float* p, int n) {
  int t = blockIdx.x * blockDim.x + threadIdx.x;
  if (t < n) p[t] = 0.f;
}

__global__ void stats_k(const float* __restrict__ Z, float* __restrict__ st, int S16) {
  __shared__ float s1[256], s2[256];
  int c = blockIdx.y, b = blockIdx.z;
  int i = blockIdx.x * 256 + threadIdx.x;
  float v = Z[((size_t)b * 16 + c) * S16 + i];
  s1[threadIdx.x] = v; s2[threadIdx.x] = v * v;
  __syncthreads();
  for (int w = 128; w > 0; w >>= 1) {
    if (threadIdx.x < w) { s1[threadIdx.x] += s1[threadIdx.x + w]; s2[threadIdx.x] += s2[threadIdx.x + w]; }
    __syncthreads();
  }
  if (threadIdx.x == 0) {
    atomicAdd(&st[(b * 16 + c) * 2 + 0], s1[0]);
    atomicAdd(&st[(b * 16 + c) * 2 + 1], s2[0]);
  }
}

// ---------------------------------------------------------------------------
// Apply InstanceNorm + ReLU in place; emit mp[c,s] = max_k(mat*dw)
// ---------------------------------------------------------------------------
__global__ void in_apply_k(float* __restrict__ Z, const float* __restrict__ st,
                           const float* __restrict__ iw, const float* __restrict__ ib,
                           const float* __restrict__ dw, float* __restrict__ mp, int S) {
  int t = blockIdx.x * blockDim.x + threadIdx.x;
  if (t >= NB * 16 * S) return;
  int s = t % S; int c = (t / S) % 16; int b = t / (16 * S);
  int S16 = S * 16;
  float cnt = (float)S16;
  float m = st[(b * 16 + c) * 2 + 0] / cnt;
  float var = st[(b * 16 + c) * 2 + 1] / cnt - m * m;
  float rs = rsqrtf(var + 1e-5f);
  float w = iw[c], bb = ib[c];
  float mx = -3.4e38f;
  size_t base = ((size_t)b * 16 + c) * S16 + (size_t)s * 16;
  size_t dbase = (size_t)b * S16 + (size_t)s * 16;
#pragma unroll
  for (int k = 0; k < 16; ++k) {
    float v = (Z[base + k] - m) * rs * w + bb;
    v = fmaxf(v, 0.f);
    Z[base + k] = v;
    mx = fmaxf(mx, v * dw[dbase + k]);
  }
  mp[((size_t)b * 16 + c) * S + s] = mx;
}

// ---------------------------------------------------------------------------
// agg[c,s,j] = sum_k feats[c, idx[s,k]] * mat3[j,s,k]   (WMMA, K=16 -> 4 steps)
// B pairs contiguous -> b64 loads.
// ---------------------------------------------------------------------------
__global__ void agg_wmma_k(const float* __restrict__ feats, int C, int N,
                           const int* __restrict__ idx, const float* __restrict__ mat3,
                           float* __restrict__ agg, int S) {
  int lane = threadIdx.x & 31;
  int half = lane >> 4, p = lane & 15;
  int s = blockIdx.x, b = blockIdx.z, cBase = blockIdx.y * 16;
  int S16 = S * 16;
  const int* ir = idx + ((size_t)b * S + s) * KNN;
  const float* fb = feats + (size_t)b * C * N;
  const float* mb = mat3 + (size_t)b * 16 * S16 + (size_t)s * 16;
  v8f acc = {0.f, 0.f, 0.f, 0.f, 0.f, 0.f, 0.f, 0.f};
  int off = half ? 2 : 0;
  int c = cBase + p;
#pragma unroll
  for (int k0 = 0; k0 < 16; k0 += 4) {
    int ka = k0 + off, kb = ka + 1;
    float a0 = (c < C) ? fb[(size_t)c * N + ir[ka]] : 0.f;
    float a1 = (c < C) ? fb[(size_t)c * N + ir[kb]] : 0.f;
    v2f Bv = *(const v2f*)(mb + (size_t)p * S16 + ka);
    v2f A = {a0, a1};
    acc = wmma_f32_k4(A, Bv, acc);
  }
#pragma unroll
  for (int dv = 0; dv < 8; ++dv) {
    int cc = cBase + half * 8 + dv;
    if (cc < C)
      agg[(((size_t)b * C + cc) * S + s) * 16 + p] = acc[dv];
  }
}

// ---------------------------------------------------------------------------
// out[o,s] = sum_{c,j} cv_w[o,c,j] * agg[c,s,j]  (WMMA GEMM, K=C*16), +BN/ReLU
// A rows contiguous -> b128 loads; B pairs contiguous -> b64 loads.
// ---------------------------------------------------------------------------
__global__ void cvw_wmma_k(const float* __restrict__ Wf, const float* __restrict__ agg,
                           const float* __restrict__ bnS, const float* __restrict__ bnB,
                           int flags, float* __restrict__ Y, int C, int Cout, int S) {
  int lane = threadIdx.x & 31;
  int half = lane >> 4, p = lane & 15;
  int sBase = blockIdx.x * 16, cBase = blockIdx.y * 16, b = blockIdx.z;
  int K = C * 16;
  const float* Wrow = Wf + (size_t)(cBase + p) * K;
  v8f acc = {0.f, 0.f, 0.f, 0.f, 0.f, 0.f, 0.f, 0.f};
  int off = half ? 2 : 0;
  for (int k0 = 0; k0 < K; k0 += 4) {
    v4f w4 = *(const v4f*)(Wrow + k0);
    int ra = k0 + off;
    int ca = ra >> 4, ja = ra & 15;   // 4-chunk never crosses a c boundary
    v2f A = half ? (v2f){w4.z, w4.w} : (v2f){w4.x, w4.y};
    v2f Bv = *(const v2f*)(agg + (((size_t)b * C + ca) * S + sBase + p) * 16 + ja);
    acc = wmma_f32_k4(A, Bv, acc);
  }
  float* Yb = Y + (size_t)b * Cout * S;
#pragma unroll
  for (int dv = 0; dv < 8; ++dv) {
    int c = cBase + half * 8 + dv;
    float v = acc[dv];
    if (flags & 2) v = v * bnS[c] + bnB[c];
    if (flags & 4) v = fmaxf(v, 0.f);
    Yb[(size_t)c * S + sBase + p] = v;
  }
}

// ---------------------------------------------------------------------------
// max-pool over neighbor indices; residual add+relu; mean; final FC
// ---------------------------------------------------------------------------
__global__ void maxpool_k(const float* __restrict__ X, int C, int N,
                          const int* __restrict__ idx, float* __restrict__ Y, int S) {
  int t = blockIdx.x * blockDim.x + threadIdx.x;
  if (t >= NB * C * S) return;
  int s = t % S; int c = (t / S) % C; int b = t / (C * S);
  float mx = -3.4e38f;
#pragma unroll
  for (int k = 0; k < KNN; ++k) {
    int n = idx[((size_t)b * S + s) * KNN + k];
    mx = fmaxf(mx, X[((size_t)b * C + c) * N + n]);
  }
  Y[((size_t)b * C + c) * S + s] = mx;
}

__global__ void addrelu_k(const float* __restrict__ H, const float* __restrict__ XS,
                          float* __restrict__ O, int n) {
  int t = blockIdx.x * blockDim.x + threadIdx.x;
  if (t < n) O[t] = fmaxf(H[t] + XS[t], 0.f);
}

__global__ void mean_k(const float* __restrict__ X, float* __restrict__ F, int C, int S) {
  int t = blockIdx.x * blockDim.x + threadIdx.x;
  if (t >= NB * C) return;
  int c = t % C, b = t / C;
  float s = 0.f;
  for (int i = 0; i < S; ++i) s += X[((size_t)b * C + c) * S + i];
  F[(size_t)b * C + c] = s / (float)S;
}

__global__ void fcout_k(const float* __restrict__ F, const float* __restrict__ W,
                        const float* __restrict__ bias, float* __restrict__ out,
                        int C, int O) {
  int t = blockIdx.x * blockDim.x + threadIdx.x;
  if (t >= NB * O) return;
  int o = t % O, b = t / O;
  float s = bias[o];
  for (int c = 0; c < C; ++c) s += F[(size_t)b * C + c] * W[(size_t)o * C + c];
  out[(size_t)b * O + o] = s;
}

// ---------------------------------------------------------------------------
// Host-side parameter structs & orchestration
// ---------------------------------------------------------------------------
struct BNP { const float *b, *s; };
struct ConvP { const float *b, *w; };
struct FkaP {
  const float *alpha, *beta, *cv_w, *fc1_w, *fc2_w, *fc3_w;
  const float *in1_b, *in1_w, *in2_b, *in2_w, *norm_radius;
};
struct BlockP {
  BNP bn0, bn1, bn2, bn_sc;
  ConvP cv0; FkaP cv1; ConvP cv2, sc;
  bool has_sc;
};
struct Ws {
  float *posT, *xT;
  int *idx0, *idxB;
  float *bufX, *bufH, *bufF, *bufO, *bufS, *bufP;
  float *pts, *dw, *zA, *zB, *zC, *mp1, *mp2, *agg, *stats, *feat;
};

static inline float* alloc_f(char*& p, size_t n) {
  float* r = (float*)p;
  p += ((n * 4 + 255) / 256) * 256;
  return r;
}

static void run_fka(hipStream_t st, const FkaP& P, const float* feats, int C, int N,
                    const float* posT, int n0, int S, const int* idx,
                    const float* bnS, const float* bnB, int flags,
                    float* out, int Cout, Ws& w) {
  int S16 = S * 16;
  geom_k<<<dim3((NB * S + 255) / 256), 256, 0, st>>>(posT, n0, idx, S, P.alpha, P.beta,
                                                     P.norm_radius, w.pts, w.dw);
  // fc1 -> zA (raw), IN1 stats, apply IN+ReLU, mp1
  mlp_wmma_k<<<dim3(S, 1, NB), 32, 0, st>>>(P.fc1_w, 3, w.pts, nullptr, 0, w.dw, 0, w.zA, S);
  zero_k<<<1, 64, 0, st>>>(w.stats, 64);
  stats_k<<<dim3(S16 / 256, 16, NB), 256, 0, st>>>(w.zA, w.stats, S16);
  in_apply_k<<<dim3((NB * 16 * S + 255) / 256), 256, 0, st>>>(w.zA, w.stats, P.in1_w,
                                                              P.in1_b, w.dw, w.mp1, S);
  // fc2 on concat(mat1, mp1) -> zB, IN2, mp2
  mlp_wmma_k<<<dim3(S, 1, NB), 32, 0, st>>>(P.fc2_w, 32, w.zA, w.mp1, 1, w.dw, 0, w.zB, S);
  zero_k<<<1, 64, 0, st>>>(w.stats, 64);
  stats_k<<<dim3(S16 / 256, 16, NB), 256, 0, st>>>(w.zB, w.stats, S16);
  in_apply_k<<<dim3((NB * 16 * S + 255) / 256), 256, 0, st>>>(w.zB, w.stats, P.in2_w,
                                                              P.in2_b, w.dw, w.mp2, S);
  // fc3 on concat(mat2, mp2), relu * dw -> mat3 (zC)
  mlp_wmma_k<<<dim3(S, 1, NB), 32, 0, st>>>(P.fc3_w, 32, w.zB, w.mp2, 1, w.dw, 1, w.zC, S);
  // aggregate gathered features against mat3
  int Ct = (C + 15) / 16;
  agg_wmma_k<<<dim3(S, Ct, NB), 32, 0, st>>>(feats, C, N, idx, w.zC, w.agg, S);
  // contract with cv_w (flattened [Cout, C*16]) + fused BN/ReLU
  cvw_wmma_k<<<dim3(S / 16, Cout / 16, NB), 32, 0, st>>>(P.cv_w, w.agg, bnS, bnB, flags,
                                                         out, C, Cout, S);
}

static void run_block(hipStream_t st, const BlockP& P, int cin, int cout, int N, int S,
                      float radius, const int* idx_reuse, Ws& w) {
  int mid = cin / 2;
  const int* idx = idx_reuse;
  if (!idx) {
    knn_k<<<dim3((S + 255) / 256, NB), 256, 0, st>>>(w.posT, N0, N, S, radius * radius,
                                                     w.idxB);
    idx = w.idxB;
  }
  // h = relu(bn0(cv0(x)))
  conv1_wmma_k<<<dim3(N / 16, mid / 16, NB), 32, 0, st>>>(P.cv0.w, P.cv0.b, P.bn0.s,
                                                          P.bn0.b, 1 | 2 | 4, w.bufX,
                                                          w.bufH, cin, mid, N);
  // h = relu(bn1(fkaconv(h)))
  run_fka(st, P.cv1, w.bufH, mid, N, w.posT, N0, S, idx, P.bn1.s, P.bn1.b, 2 | 4, w.bufF,
          mid, w);
  // h = bn2(cv2(h))
  conv1_wmma_k<<<dim3(S / 16, cout / 16, NB), 32, 0, st>>>(P.cv2.w, P.cv2.b, P.bn2.s,
                                                           P.bn2.b, 1 | 2, w.bufF, w.bufO,
                                                           mid, cout, S);
  // shortcut
  const float* xs;
  if (P.has_sc) {
    conv1_wmma_k<<<dim3(N / 16, cout / 16, NB), 32, 0, st>>>(P.sc.w, P.sc.b, P.bn_sc.s,
                                                             P.bn_sc.b, 1 | 2, w.bufX,
                                                             w.bufS, cin, cout, N);
    if (N != S) {
      maxpool_k<<<dim3((NB * cout * S + 255) / 256), 256, 0, st>>>(w.bufS, cout, N, idx,
                                                                   w.bufP, S);
      xs = w.bufP;
    } else {
      xs = w.bufS;
    }
  } else {
    xs = w.bufX;
  }
  int n = NB * cout * S;
  addrelu_k<<<dim3((n + 255) / 256), 256, 0, st>>>(w.bufO, xs, w.bufX, n);
}

// ---------------------------------------------------------------------------
extern "C" void kernel_launch(void* const* d_in, const int* in_sizes, int n_in,
                              void* d_out, int out_size, void* d_ws, size_t ws_size,
                              hipStream_t stream) {
  (void)in_sizes; (void)n_in; (void)out_size; (void)ws_size;
  const float* pos = (const float*)d_in[0];
  const float* x = (const float*)d_in[1];

  // --- unpack params (jax pytree: nested dicts flattened in sorted-key order)
  int ti = 2;
  auto nx = [&]() { return (const float*)d_in[ti++]; };
  auto rdBN = [&]() { BNP p; p.b = nx(); p.s = nx(); return p; };
  auto rdCv = [&]() { ConvP p; p.b = nx(); p.w = nx(); return p; };
  auto rdFka = [&]() {
    FkaP p;
    p.alpha = nx(); p.beta = nx(); p.cv_w = nx();
    p.fc1_w = nx(); p.fc2_w = nx(); p.fc3_w = nx();
    p.in1_b = nx(); p.in1_w = nx(); p.in2_b = nx(); p.in2_w = nx();
    p.norm_radius = nx();
    return p;
  };
  auto rdBlk = [&](bool sc) {
    BlockP p{};
    p.bn0 = rdBN(); p.bn1 = rdBN(); p.bn2 = rdBN();
    if (sc) p.bn_sc = rdBN();
    p.cv0 = rdCv(); p.cv1 = rdFka(); p.cv2 = rdCv();
    if (sc) p.sc = rdCv();
    p.has_sc = sc;
    return p;
  };
  BlockP b01 = rdBlk(false), b10 = rdBlk(true), b11 = rdBlk(false), b20 = rdBlk(true),
         b21 = rdBlk(false), b30 = rdBlk(true), b31 = rdBlk(false), b40 = rdBlk(true),
         b41 = rdBlk(false);
  BNP bn0 = rdBN();
  FkaP cv0 = rdFka();
  ConvP fcout = rdCv();

  // --- workspace bump allocation
  char* p = (char*)d_ws;
  Ws w{};
  w.posT = alloc_f(p, (size_t)NB * 3 * N0);
  w.xT = alloc_f(p, (size_t)NB * 3 * N0);
  w.idx0 = (int*)alloc_f(p, (size_t)NB * N0 * KNN);
  w.idxB = (int*)alloc_f(p, (size_t)NB * 2048 * KNN);
  w.bufX = alloc_f(p, (size_t)NB * 64 * 8192);
  w.bufH = alloc_f(p, (size_t)NB * 32 * 8192);
  w.bufF = alloc_f(p, (size_t)NB * 32 * 8192);
  w.bufO = alloc_f(p, (size_t)NB * 64 * 8192);
  w.bufS = alloc_f(p, (size_t)NB * 128 * 8192);
  w.bufP = alloc_f(p, (size_t)NB * 128 * 2048);
  w.pts = alloc_f(p, (size_t)NB * 3 * 8192 * 16);
  w.dw = alloc_f(p, (size_t)NB * 8192 * 16);
  w.zA = alloc_f(p, (size_t)NB * 16 * 8192 * 16);
  w.zB = alloc_f(p, (size_t)NB * 16 * 8192 * 16);
  w.zC = alloc_f(p, (size_t)NB * 16 * 8192 * 16);
  w.mp1 = alloc_f(p, (size_t)NB * 16 * 8192);
  w.mp2 = alloc_f(p, (size_t)NB * 16 * 8192);
  w.agg = alloc_f(p, (size_t)NB * 32 * 8192 * 16);
  w.stats = alloc_f(p, 64);
  w.feat = alloc_f(p, (size_t)NB * 1024);

  const float R0f = 2.5f * 0.06f;

  // --- network
  transpose_k<<<(NB * N0 + 255) / 256, 256, 0, stream>>>(pos, w.posT, N0);
  transpose_k<<<(NB * N0 + 255) / 256, 256, 0, stream>>>(x, w.xT, N0);

  knn_k<<<dim3((N0 + 255) / 256, NB), 256, 0, stream>>>(w.posT, N0, N0, N0, R0f * R0f,
                                                        w.idx0);
  // x0 = relu(bn0(fkaconv(cv0, x, pos, pos, ids0)))
  run_fka(stream, cv0, w.xT, 3, N0, w.posT, N0, N0, w.idx0, bn0.s, bn0.b, 2 | 4, w.bufX,
          64, w);
  run_block(stream, b01, 64, 64, 8192, 8192, R0f, w.idx0, w);       // same support/radius as ids0
  run_block(stream, b10, 64, 128, 8192, 2048, R0f, nullptr, w);
  run_block(stream, b11, 128, 128, 2048, 2048, 2 * R0f, nullptr, w);
  run_block(stream, b20, 128, 256, 2048, 512, 2 * R0f, nullptr, w);
  run_block(stream, b21, 256, 256, 512, 512, 4 * R0f, nullptr, w);
  run_block(stream, b30, 256, 512, 512, 128, 4 * R0f, nullptr, w);
  run_block(stream, b31, 512, 512, 128, 128, 8 * R0f, nullptr, w);
  run_block(stream, b40, 512, 1024, 128, 32, 8 * R0f, nullptr, w);
  run_block(stream, b41, 1024, 1024, 32, 32, 16 * R0f, nullptr, w);

  mean_k<<<(NB * 1024 + 255) / 256, 256, 0, stream>>>(w.bufX, w.feat, 1024, 32);
  fcout_k<<<1, 128, 0, stream>>>(w.feat, fcout.w, fcout.b, (float*)d_out, 1024, 40);
}